// PackBits_61289183314094
// MI455X (gfx1250) — compile-verified
//
#include <hip/hip_runtime.h>
#include <stdint.h>

// PackBits: out[w] bit i = signbit(x[w*32 + (i^7)])   (numpy packbits MSB-first
// within byte, then little-endian uint32 view == XOR-7 bit swizzle).
//
// Wave32-native strategy: lane i loads element (i^7) of word w, predicate
// "(int)bits < 0" (exact signbit semantics incl. -0.0 / NaN sign), and a single
// v_cmp -> ballot_w32 produces the finished 32-bit packed word for the whole
// wave. 32 ballots per wave-tile = 32 words, kept one-per-lane, stored as one
// coalesced 128B global_store_b32. Memory-bound: 512MB in + 16MB out ~= 23us
// at 23.3 TB/s; NT hints keep the single-pass stream out of the 192MB L2.

__global__ __launch_bounds__(256) void PackBits_61289183314094_kernel(
    const uint32_t* __restrict__ xu,   // float32 bits viewed as uint32
    uint32_t* __restrict__ out,        // packed words
    int nwords) {
  const int lane    = (int)(threadIdx.x & 31u);
  const int waveId  = (int)((blockIdx.x * blockDim.x + threadIdx.x) >> 5);
  const int nwaves  = (int)((gridDim.x * blockDim.x) >> 5);
  const int srcLane = lane ^ 7;  // byte-reversed bit order within each byte

  // Grid-stride over tiles of 32 words (1024 elements) per wave.
  // Loop bound is wave-uniform, so EXEC is all-ones at every ballot.
  for (int base = waveId * 32; base < nwords; base += nwaves * 32) {
    uint32_t myword = 0;
#pragma unroll
    for (int k = 0; k < 32; ++k) {
      // Word (base+k): lane i reads element (i^7). Contiguous 128B per load.
      const size_t idx = ((size_t)(base + k) << 5) + (size_t)srcLane;
      const uint32_t bits = __builtin_nontemporal_load(xu + idx);
      // signbit == MSB set == negative as signed int (handles -0.0, NaNs).
      const uint32_t ballot = __builtin_amdgcn_ballot_w32((int32_t)bits < 0);
      // Ballot bit i = sign(x[w*32 + (i^7)]) -> exactly the packed word.
      myword = (lane == k) ? ballot : myword;
    }
    // 32 words accumulated one-per-lane -> one coalesced 128B store.
    __builtin_nontemporal_store(myword, out + base + lane);
  }
}

extern "C" void kernel_launch(void* const* d_in, const int* in_sizes, int n_in,
                              void* d_out, int out_size, void* d_ws, size_t ws_size,
                              hipStream_t stream) {
  (void)n_in; (void)d_ws; (void)ws_size;
  const uint32_t* xu = (const uint32_t*)d_in[0];  // float32 input, bit view
  uint32_t* out = (uint32_t*)d_out;               // uint32 output words

  const int nwords = out_size;               // = in_sizes[0] / 32
  const int threads = 256;                   // 8 wave32s per block
  const int wavesPerBlock = threads / 32;
  const int tiles = (nwords + 31) / 32;      // one 32-word tile per wave
  int blocks = (tiles + wavesPerBlock - 1) / wavesPerBlock;
  if (blocks < 1) blocks = 1;
  if (blocks > 65535 * 16) blocks = 65535 * 16;  // grid-stride covers the rest

  PackBits_61289183314094_kernel<<<blocks, threads, 0, stream>>>(xu, out, nwords);
}